// TransducerMultiHeadModelPhaseBatch_28389733827071
// MI455X (gfx1250) — compile-verified
//
#include <hip/hip_runtime.h>
#include <math.h>

#define NBINS 314
#define HDIM  256
#define DDIM  197

typedef float v2f __attribute__((ext_vector_type(2)));
typedef float v8f __attribute__((ext_vector_type(8)));

__device__ __forceinline__ float gelu_exact(float x) {
    return 0.5f * x * (1.0f + erff(x * 0.7071067811865475f));
}
__device__ __forceinline__ v2f nt_load2(const float* p) {
    return __builtin_nontemporal_load((const v2f*)p);
}
__device__ __forceinline__ float nt_load1(const float* p) {
    return __builtin_nontemporal_load(p);
}

// A-operand for V_WMMA_F32_16X16X4_F32 loaded directly in ISA layout:
// 16x4 f32 A: lanes 0-15 = M rows with (K0,K1) in (v0,v1); lanes 16-31 same
// rows with (K2,K3). So lane L reads b64 at row*256 + k + (L>=16 ? 2 : 0).
// B operand = x replicated across all 16 N columns -> every D column = y[m].

// One 16-row tile (rows clamped to maxrow-1), W row-major ld=256.
__device__ __forceinline__ v8f gemv_tile_wmma(const float* __restrict__ wmat,
                                              int r0, int maxrow,
                                              const float* xs, int lane) {
    v8f acc;
#pragma unroll
    for (int i = 0; i < 8; ++i) acc[i] = 0.f;
    int row = r0 + (lane & 15);
    if (row >= maxrow) row = maxrow - 1;
    const int koff = (lane >> 4) * 2;
    const float* wp = wmat + (size_t)row * 256 + koff;
#pragma unroll 2
    for (int k0 = 0; k0 < 256; k0 += 8) {
        v2f a1 = nt_load2(wp + k0);          // k-group [k0,   k0+3]
        v2f a2 = nt_load2(wp + k0 + 4);      // k-group [k0+4, k0+7]
        v2f b1 = *(const v2f*)(xs + k0 + koff);
        v2f b2 = *(const v2f*)(xs + k0 + 4 + koff);
        acc = __builtin_amdgcn_wmma_f32_16x16x4_f32(false, a1, false, b1,
                                                    (short)0, acc, false, false);
        acc = __builtin_amdgcn_wmma_f32_16x16x4_f32(false, a2, false, b2,
                                                    (short)0, acc, false, false);
    }
    return acc;
}

// Two 16-row tiles fused in one k-loop; B-operand LDS reads shared.
__device__ __forceinline__ void gemv_tile2_wmma(const float* __restrict__ wmat,
                                                int r0a, int r0b,
                                                const float* xs, int lane,
                                                v8f& accA, v8f& accB) {
#pragma unroll
    for (int i = 0; i < 8; ++i) { accA[i] = 0.f; accB[i] = 0.f; }
    const int koff = (lane >> 4) * 2;
    const float* wpa = wmat + (size_t)(r0a + (lane & 15)) * 256 + koff;
    const float* wpb = wmat + (size_t)(r0b + (lane & 15)) * 256 + koff;
#pragma unroll 2
    for (int k0 = 0; k0 < 256; k0 += 8) {
        v2f a1a = nt_load2(wpa + k0);
        v2f a2a = nt_load2(wpa + k0 + 4);
        v2f a1b = nt_load2(wpb + k0);
        v2f a2b = nt_load2(wpb + k0 + 4);
        v2f b1  = *(const v2f*)(xs + k0 + koff);
        v2f b2  = *(const v2f*)(xs + k0 + 4 + koff);
        accA = __builtin_amdgcn_wmma_f32_16x16x4_f32(false, a1a, false, b1,
                                                     (short)0, accA, false, false);
        accA = __builtin_amdgcn_wmma_f32_16x16x4_f32(false, a2a, false, b2,
                                                     (short)0, accA, false, false);
        accB = __builtin_amdgcn_wmma_f32_16x16x4_f32(false, a1b, false, b1,
                                                     (short)0, accB, false, false);
        accB = __builtin_amdgcn_wmma_f32_16x16x4_f32(false, a2b, false, b2,
                                                     (short)0, accB, false, false);
    }
}

__global__ __launch_bounds__(256)
void tmhm_phase_kernel(const float* __restrict__ target_xyz,
                       const float* __restrict__ trans_xyz,
                       const float* __restrict__ out_inter,
                       const float* __restrict__ in_inter,
                       const float* __restrict__ hu_line,
                       const float* __restrict__ ln_w,
                       const float* __restrict__ ln_b,
                       const float* __restrict__ fc1_w, const float* __restrict__ fc1_b,
                       const float* __restrict__ fc2_w, const float* __restrict__ fc2_b,
                       const float* __restrict__ fc3_w, const float* __restrict__ fc3_b,
                       const float* __restrict__ out_w, const float* __restrict__ out_b,
                       float* __restrict__ out_phi, float* __restrict__ out_logits) {
    const int b    = blockIdx.x;
    const int t    = threadIdx.x;
    const int lane = t & 31;
    const int wid  = t >> 5;

    __shared__ float sA[256];
    __shared__ float sB[256];
    __shared__ float sLog[320];
    __shared__ float sR1[256];
    __shared__ float sR2[256];

    // ---------------- stage 1: geo features into sA[0..196] ----------------
    sR1[t] = hu_line[(size_t)b * 256 + t];
    if (t < 192) {
        int i = t >> 5, j = t & 31, c = i - (i >= 3 ? 3 : 0);
        float comp = (i < 3) ? trans_xyz[b * 3 + c] : target_xyz[b * 3 + c];
        float off  = (c == 2) ? 0.092f : 0.132f;
        float val  = comp * 0.001f - off;
        float freq = (float)(1 << (j & 15)) * 6.283185307179586f;
        float ang  = val * freq;
        sA[t] = (j < 16) ? sinf(ang) : cosf(ang);
    } else if (t >= 197) {
        sA[t] = 0.f;
    }
    __syncthreads();
    for (int s = 128; s > 0; s >>= 1) { if (t < s) sR1[t] += sR1[t + s]; __syncthreads(); }
    if (t == 0) {
        float tx = target_xyz[b*3], ty = target_xyz[b*3+1], tz = target_xyz[b*3+2];
        float rx = trans_xyz[b*3],  ry = trans_xyz[b*3+1],  rz = trans_xyz[b*3+2];
        float ox = out_inter[b*3],  oy = out_inter[b*3+1],  oz = out_inter[b*3+2];
        float ix = in_inter[b*3],   iy = in_inter[b*3+1],   iz = in_inter[b*3+2];
        float d_o = sqrtf((ox-rx)*(ox-rx)+(oy-ry)*(oy-ry)+(oz-rz)*(oz-rz)) * 0.001f;
        float d_t = sqrtf((ox-ix)*(ox-ix)+(oy-iy)*(oy-iy)+(oz-iz)*(oz-iz)) * 0.001f;
        float d_i = sqrtf((ix-tx)*(ix-tx)+(iy-ty)*(iy-ty)+(iz-tz)*(iz-tz)) * 0.001f;
        float tof = (d_o / 1500.f + d_t / 2262.f + d_i / 1500.f) * 1000.f;
        sA[192] = d_o; sA[193] = d_t; sA[194] = d_i; sA[195] = tof;
        sA[196] = sR1[0] * (1.f / 256.f);
    }
    __syncthreads();

    // ---------------- layernorm (per-sample affine) into sB ----------------
    float gv = (t < DDIM) ? sA[t] : 0.f;
    sR1[t] = gv; sR2[t] = gv * gv;
    __syncthreads();
    for (int s = 128; s > 0; s >>= 1) {
        if (t < s) { sR1[t] += sR1[t + s]; sR2[t] += sR2[t + s]; }
        __syncthreads();
    }
    float mu   = sR1[0] * (1.f / 197.f);
    float var  = sR2[0] * (1.f / 197.f) - mu * mu;
    float rstd = rsqrtf(var + 1e-5f);
    sB[t] = (t < DDIM)
          ? ((gv - mu) * rstd * ln_w[(size_t)b * DDIM + t] + ln_b[(size_t)b * DDIM + t])
          : 0.f;
    __syncthreads();

    // ---------------- fc1: [256 x 197] GEMV, wave-per-row, coalesced b32 ----
    {
        float xg[7];
#pragma unroll
        for (int i = 0; i < 7; ++i) xg[i] = sB[lane + 32 * i]; // sB zero-padded
        const float* w1 = fc1_w + (size_t)b * HDIM * DDIM;
        for (int r = wid * 32; r < wid * 32 + 32; ++r) {
            const float* wr = w1 + (size_t)r * DDIM;
            float acc = 0.f;
#pragma unroll
            for (int i = 0; i < 6; ++i) acc += nt_load1(wr + lane + 32 * i) * xg[i];
            if (lane < 5) acc += nt_load1(wr + 192 + lane) * xg[6];
#pragma unroll
            for (int off = 16; off >= 1; off >>= 1) acc += __shfl_xor(acc, off, 32);
            if (lane == 0) sA[r] = gelu_exact(acc + fc1_b[(size_t)b * HDIM + r]);
        }
    }
    __syncthreads();

    // ---------------- fc2: WMMA f32 GEMV (2 fused tiles), sA -> sB ---------
    {
        const float* w2 = fc2_w + (size_t)b * HDIM * HDIM;
        v8f accA, accB;
        gemv_tile2_wmma(w2, wid * 16, wid * 16 + 128, sA, lane, accA, accB);
        if ((lane & 15) == 0) {
            int rb = wid * 16 + (lane >> 4) * 8;
#pragma unroll
            for (int m = 0; m < 8; ++m) {
                sB[rb + m]       = gelu_exact(accA[m] + fc2_b[(size_t)b * HDIM + rb + m]);
                sB[rb + 128 + m] = gelu_exact(accB[m] + fc2_b[(size_t)b * HDIM + rb + 128 + m]);
            }
        }
    }
    __syncthreads();

    // ---------------- fc3: WMMA f32 GEMV (2 fused tiles), sB -> sA ---------
    {
        const float* w3 = fc3_w + (size_t)b * HDIM * HDIM;
        v8f accA, accB;
        gemv_tile2_wmma(w3, wid * 16, wid * 16 + 128, sB, lane, accA, accB);
        if ((lane & 15) == 0) {
            int rb = wid * 16 + (lane >> 4) * 8;
#pragma unroll
            for (int m = 0; m < 8; ++m) {
                sA[rb + m]       = gelu_exact(accA[m] + fc3_b[(size_t)b * HDIM + rb + m]);
                sA[rb + 128 + m] = gelu_exact(accB[m] + fc3_b[(size_t)b * HDIM + rb + 128 + m]);
            }
        }
    }
    __syncthreads();

    // ---------------- out: [314 x 256] WMMA GEMV -> logits -----------------
    {
        const float* wo = out_w + (size_t)b * NBINS * HDIM;
        // tiles 0..15 (rows 0..255): fused pairs, no clamping needed
        {
            v8f accA, accB;
            gemv_tile2_wmma(wo, wid * 16, wid * 16 + 128, sA, lane, accA, accB);
            if ((lane & 15) == 0) {
                int rb = wid * 16 + (lane >> 4) * 8;
#pragma unroll
                for (int m = 0; m < 8; ++m) {
                    float va = accA[m] + out_b[(size_t)b * NBINS + rb + m];
                    float vb = accB[m] + out_b[(size_t)b * NBINS + rb + 128 + m];
                    sLog[rb + m]       = va;
                    sLog[rb + 128 + m] = vb;
                    out_logits[(size_t)b * NBINS + rb + m]       = va;
                    out_logits[(size_t)b * NBINS + rb + 128 + m] = vb;
                }
            }
        }
        // tiles 16..19 (rows 256..319, clamped to 313): waves 0..3
        if (wid < 4) {
            int r0 = 256 + wid * 16;
            v8f acc = gemv_tile_wmma(wo, r0, NBINS, sA, lane);
            if ((lane & 15) == 0) {
                int rb = r0 + (lane >> 4) * 8;
#pragma unroll
                for (int m = 0; m < 8; ++m) {
                    int r = rb + m;
                    if (r < NBINS) {
                        float v = acc[m] + out_b[(size_t)b * NBINS + r];
                        sLog[r] = v;
                        out_logits[(size_t)b * NBINS + r] = v;
                    }
                }
            }
        }
    }
    __syncthreads();

    // ---------------- softmax + circular mean (Z cancels in atan2) ---------
    float lm = -3.4e38f;
    for (int k = t; k < NBINS; k += 256) lm = fmaxf(lm, sLog[k]);
    sR1[t] = lm;
    __syncthreads();
    for (int s = 128; s > 0; s >>= 1) { if (t < s) sR1[t] = fmaxf(sR1[t], sR1[t + s]); __syncthreads(); }
    float mx = sR1[0];
    __syncthreads();
    float ssin = 0.f, scos = 0.f;
    for (int k = t; k < NBINS; k += 256) {
        float e   = expf(sLog[k] - mx);
        float ctr = -3.14159265358979f + 6.283185307179586f * ((float)k + 0.5f) * (1.0f / 314.0f);
        ssin += e * sinf(ctr);
        scos += e * cosf(ctr);
    }
    sR1[t] = ssin; sR2[t] = scos;
    __syncthreads();
    for (int s = 128; s > 0; s >>= 1) {
        if (t < s) { sR1[t] += sR1[t + s]; sR2[t] += sR2[t + s]; }
        __syncthreads();
    }
    if (t == 0) out_phi[b] = atan2f(sR1[0], sR2[0]);
}

extern "C" void kernel_launch(void* const* d_in, const int* in_sizes, int n_in,
                              void* d_out, int out_size, void* d_ws, size_t ws_size,
                              hipStream_t stream) {
    const float* target_xyz = (const float*)d_in[0];
    const float* trans_xyz  = (const float*)d_in[1];
    const float* out_inter  = (const float*)d_in[2];
    const float* in_inter   = (const float*)d_in[3];
    const float* hu_line    = (const float*)d_in[4];
    const float* ln_w       = (const float*)d_in[5];
    const float* ln_b       = (const float*)d_in[6];
    const float* fc1_w      = (const float*)d_in[7];
    const float* fc1_b      = (const float*)d_in[8];
    const float* fc2_w      = (const float*)d_in[9];
    const float* fc2_b      = (const float*)d_in[10];
    const float* fc3_w      = (const float*)d_in[11];
    const float* fc3_b      = (const float*)d_in[12];
    const float* out_w      = (const float*)d_in[13];
    const float* out_b      = (const float*)d_in[14];

    const int B = in_sizes[0] / 3;            // 2048
    float* out_phi    = (float*)d_out;        // [B] first, then logits [B,314]
    float* out_logits = (float*)d_out + B;

    hipLaunchKernelGGL(tmhm_phase_kernel, dim3(B), dim3(256), 0, stream,
                       target_xyz, trans_xyz, out_inter, in_inter, hu_line,
                       ln_w, ln_b, fc1_w, fc1_b, fc2_w, fc2_b, fc3_w, fc3_b,
                       out_w, out_b, out_phi, out_logits);
}